// SquareAUCLoss_mine_89962384982115
// MI455X (gfx1250) — compile-verified
//
#include <hip/hip_runtime.h>
#include <cstdint>

// ---------------------------------------------------------------------------
// SquareAUCLoss forward for MI455X (gfx1250).
// Memory-bound: streams pred[32768,1000] (131 MB) once -> ~5.6us @ 23.3TB/s.
// Main kernel stages pred through LDS with CDNA5 async global->LDS copies
// (double buffered, ASYNCcnt-tracked) so HBM stays saturated with few waves.
// ---------------------------------------------------------------------------

#define AS1 __attribute__((address_space(1)))
#define AS3 __attribute__((address_space(3)))

#ifndef __has_builtin
#define __has_builtin(x) 0
#endif

#if __has_builtin(__builtin_amdgcn_global_load_async_to_lds_b128)
#define HAVE_ASYNC_LDS 1
#endif

#ifdef HAVE_ASYNC_LDS
#if __has_builtin(__builtin_amdgcn_s_wait_asynccnt)
#define WAIT_ASYNC(n) __builtin_amdgcn_s_wait_asynccnt(n)
#else
#define WAIT_ASYNC(n) asm volatile("s_wait_asynccnt %0" ::"i"(n) : "memory")
#endif
#else
#define WAIT_ASYNC(n) ((void)0)
#endif

typedef int v4i __attribute__((ext_vector_type(4)));

__device__ __forceinline__ void copy16_to_lds(const float* g, float* l) {
#ifdef HAVE_ASYNC_LDS
  // Builtin signature (probe-confirmed): (v4i AS1*, v4i AS3*, imm, imm).
  typedef AS1 v4i* gp_t;
  typedef AS3 v4i* lp_t;
  // integer round-trips: generic->AS1 keeps the 64-bit VA; generic LDS flat
  // address has the LDS byte offset in bits [31:0] (ISA aperture rule),
  // AS3 pointers are 32-bit LDS offsets.
  gp_t gp = (gp_t)(unsigned long long)(const void*)g;
  lp_t lp = (lp_t)(unsigned int)(unsigned long long)(void*)l;
  __builtin_amdgcn_global_load_async_to_lds_b128(gp, lp, 0, 0);
#else
  *(float4*)l = *(const float4*)g;
#endif
}

// ---------------- problem constants (match reference shapes) ---------------
constexpr int kC      = 1000;   // num classes
constexpr int kCPad   = 1024;
constexpr int kTPB    = 256;    // 8 waves (wave32)
constexpr int kRows   = 128;    // rows per block (B=32768 -> 256 blocks)
constexpr int kChRows = 4;      // rows per LDS chunk
constexpr int kNCh    = kRows / kChRows;  // 32 chunks
constexpr int kBlkPerRow = kC / 4;        // 250 x b128 per row

// workspace layout (in 4-byte words)
//  [0,1024)        Ncnt (int)
//  [1024]          existCnt (int)
//  [2048,3072)     invN (float)
//  [3072,4096)     T1 (float)
//  [4096,5120)     T2 (float)
//  [5120,7168)     F  (float, A1/A2 interleaved: F[2c]=A1, F[2c+1]=A2)
//  [7168, ...)     partial (float, nMainBlocks * 2048)

__global__ void zero_kernel(unsigned int* p, int n) {
  int i = blockIdx.x * blockDim.x + threadIdx.x;
  if (i < n) p[i] = 0u;
}

// Per-row gather: N[c], T1[c]=sum_pos p, T2[c]=sum_pos p^2.
__global__ void row_gather_kernel(const float* __restrict__ pred,
                                  const int* __restrict__ tgt,
                                  int* __restrict__ Ncnt,
                                  float* __restrict__ T1,
                                  float* __restrict__ T2, int B, int C) {
  int b = blockIdx.x * blockDim.x + threadIdx.x;
  if (b >= B) return;
  int c = tgt[b];
  float p = pred[(size_t)b * C + c];
  atomicAdd(&Ncnt[c], 1);
  atomicAdd(&T1[c], p);
  atomicAdd(&T2[c], p * p);
}

__global__ void invn_kernel(const int* __restrict__ Ncnt,
                            float* __restrict__ invN,
                            int* __restrict__ existCnt, int C) {
  int c = threadIdx.x;
  if (c < C) {
    int n = Ncnt[c];
    invN[c] = (n > 0) ? (1.0f / (float)n) : 0.0f;
    if (n > 0) atomicAdd(existCnt, 1);
  }
}

// Main streamer: per-class weighted column sums A1 = sum w*p, A2 = sum w*p^2.
// Each block: 128 rows x 1000 cols, 4-row chunks double-buffered in LDS via
// async global->LDS. thread t owns columns [4t, 4t+4). Assumes B % 128 == 0
// (reference: B = 32768).
__global__ __launch_bounds__(kTPB) void auc_main_kernel(
    const float* __restrict__ pred, const int* __restrict__ tgt,
    const float* __restrict__ invN, float* __restrict__ partial) {
  __shared__ float buf[2][kChRows][kC];  // 32000 B
  __shared__ float wrow[kRows];          // 512 B

  const int t  = threadIdx.x;
  const int b0 = blockIdx.x * kRows;

  // Preload per-row weights w[b] = invN[target[b]] (one gather per row).
  if (t < kRows) wrow[t] = invN[tgt[b0 + t]];

  // Issue one 4-row chunk into LDS buffer bi. Exactly 4 async instructions
  // per wave per chunk (threads 250..255 masked -> wave 7 partial EXEC).
  auto issue_chunk = [&](int ch, int bi) {
    const float* gbase = pred + (size_t)(b0 + ch * kChRows) * kC;
    if (t < kBlkPerRow) {
#pragma unroll
      for (int r = 0; r < kChRows; ++r) {
        copy16_to_lds(gbase + (size_t)r * kC + t * 4, &buf[bi][r][t * 4]);
      }
    }
  };

  float a1[4] = {0.f, 0.f, 0.f, 0.f};
  float a2[4] = {0.f, 0.f, 0.f, 0.f};

  issue_chunk(0, 0);
  for (int ch = 0; ch < kNCh; ++ch) {
    const int bi = ch & 1;
    if (ch + 1 < kNCh) {
      issue_chunk(ch + 1, bi ^ 1);  // prefetch next chunk
      WAIT_ASYNC(4);                // chunk ch (in-order) complete, next in flight
    } else {
      WAIT_ASYNC(0);
    }
    __syncthreads();  // all waves' chunk-ch data (and wrow) visible

    const int col = t * 4;
    if (col < kC) {
#pragma unroll
      for (int r = 0; r < kChRows; ++r) {
        const float w = wrow[ch * kChRows + r];
        const float4 p = *(const float4*)&buf[bi][r][col];
        float wp;
        wp = w * p.x; a1[0] += wp; a2[0] = fmaf(wp, p.x, a2[0]);
        wp = w * p.y; a1[1] += wp; a2[1] = fmaf(wp, p.y, a2[1]);
        wp = w * p.z; a1[2] += wp; a2[2] = fmaf(wp, p.z, a2[2]);
        wp = w * p.w; a1[3] += wp; a2[3] = fmaf(wp, p.w, a2[3]);
      }
    }
    __syncthreads();  // buffer bi free for reuse
  }

  // Per-block partials (no global float atomics: 256 blocks -> tree reduce).
  float* out = partial + (size_t)blockIdx.x * (2 * kCPad);
  const int col = t * 4;
  if (col < kC) {
#pragma unroll
    for (int j = 0; j < 4; ++j) {
      out[2 * (col + j) + 0] = a1[j];
      out[2 * (col + j) + 1] = a2[j];
    }
  }
}

__global__ void reduce_partials_kernel(const float* __restrict__ partial,
                                       float* __restrict__ F, int nblocks) {
  int i = blockIdx.x * blockDim.x + threadIdx.x;  // [0, 2048)
  float s = 0.f;
  for (int b = 0; b < nblocks; ++b) s += partial[(size_t)b * (2 * kCPad) + i];
  F[i] = s;
}

__global__ __launch_bounds__(1024) void final_kernel(
    const int* __restrict__ Ncnt, const float* __restrict__ invN,
    const float* __restrict__ T1, const float* __restrict__ T2,
    const float* __restrict__ F, const int* __restrict__ existCnt,
    float* __restrict__ out, int C) {
  __shared__ float red[1024];
  const int c = threadIdx.x;
  const float exist = (float)(*existCnt);
  float v = 0.f;
  if (c < C) {
    int n = Ncnt[c];
    if (n > 0) {
      float inv = invN[c];
      float s0 = exist - 1.0f;              // S0 = Wtot - 1, Wtot == exist
      float t1 = T1[c], t2 = T2[c];
      float s1 = F[2 * c + 0] - inv * t1;   // S1 = A1 - invN*T1
      float s2 = F[2 * c + 1] - inv * t2;   // S2 = A2 - invN*T2
      float pair = (float)n * (s0 + 2.f * s1 + s2) -
                   2.f * t1 * (s0 + s1) + t2 * s0;
      float denom = fmaxf(exist - 1.0f, 1.0f);
      v = pair * inv / denom;
    }
  }
  red[threadIdx.x] = v;
  __syncthreads();
  for (int s = 512; s > 0; s >>= 1) {
    if (threadIdx.x < s) red[threadIdx.x] += red[threadIdx.x + s];
    __syncthreads();
  }
  if (threadIdx.x == 0) out[0] = red[0] / exist;
}

extern "C" void kernel_launch(void* const* d_in, const int* in_sizes, int n_in,
                              void* d_out, int out_size, void* d_ws,
                              size_t ws_size, hipStream_t stream) {
  const float* pred = (const float*)d_in[0];
  const int* tgt = (const int*)d_in[1];
  const int B = in_sizes[1];              // 32768
  const int C = in_sizes[0] / B;          // 1000

  int* wsI = (int*)d_ws;
  float* wsF = (float*)d_ws;
  int* Ncnt = wsI;                        // [0,1024)
  int* exist = wsI + 1024;                // [1024]
  float* invN = wsF + 2048;
  float* T1 = wsF + 3072;
  float* T2 = wsF + 4096;
  float* F = wsF + 5120;                  // 2048 floats
  float* partial = wsF + 7168;            // nMain * 2048 floats

  const int nMain = B / kRows;            // 256 (B % 128 == 0)

  zero_kernel<<<(5120 + 255) / 256, 256, 0, stream>>>((unsigned int*)d_ws, 5120);
  row_gather_kernel<<<(B + 255) / 256, 256, 0, stream>>>(pred, tgt, Ncnt, T1,
                                                         T2, B, C);
  invn_kernel<<<1, 1024, 0, stream>>>(Ncnt, invN, exist, C);
  auc_main_kernel<<<nMain, kTPB, 0, stream>>>(pred, tgt, invN, partial);
  reduce_partials_kernel<<<(2 * kCPad) / 256, 256, 0, stream>>>(partial, F,
                                                                nMain);
  final_kernel<<<1, 1024, 0, stream>>>(Ncnt, invN, T1, T2, F, exist,
                                       (float*)d_out, C);
}